// PhysicsInformedNN_42992622633174
// MI455X (gfx1250) — compile-verified
//
#include <hip/hip_runtime.h>

typedef __attribute__((ext_vector_type(2))) float v2f;
typedef __attribute__((ext_vector_type(8))) float v8f;

#define NPTS   16      // points per workgroup (WMMA M)
#define WIDTH  128     // hidden width
#define HP     132     // padded LDS pitch for h rows (floats) -> avoids 64-bank conflicts
#define NCH    20      // jet channels: 1 val + 3 grad + 6 hess + 10 third

#define H_OFF   0
#define W_OFF   (NCH * NPTS * HP)                 // 42240
#define ZF_OFF  (W_OFF + WIDTH * WIDTH)           // + 16384
#define LDS_FLOATS (ZF_OFF + NPTS * NCH)          // + 320 = 58944 floats = 235776 B

struct Params {
    const float* W[9];
    const float* b[9];
    const float* x;
    const float* y;
    const float* t;
    const float* lam1;
    const float* lam2;
    float* out;
    int n;
};

// tanh jet chain rule (Faa di Bruno to 3rd order), pointwise over 20 channels.
__device__ __forceinline__ void tanh_chain(const float z[NCH], float o[NCH]) {
    const int PA[6] = {0, 0, 0, 1, 1, 2};
    const int PB[6] = {0, 1, 2, 1, 2, 2};
    const int TA[10] = {0, 0, 0, 0, 0, 0, 1, 1, 1, 2};
    const int TB[10] = {0, 0, 0, 1, 1, 2, 1, 1, 2, 2};
    const int TC[10] = {0, 1, 2, 1, 2, 2, 1, 2, 2, 2};
    const int HIDX[3][3] = {{0, 1, 2}, {1, 3, 4}, {2, 4, 5}};

    float s  = tanhf(z[0]);
    float s1 = 1.0f - s * s;               // tanh'
    float s2 = -2.0f * s * s1;             // tanh''
    float s3 = -2.0f * (s1 * s1 + s * s2); // tanh'''

    float g[3], hz[6];
    o[0] = s;
    #pragma unroll
    for (int a = 0; a < 3; ++a) { g[a] = z[1 + a]; o[1 + a] = s1 * g[a]; }
    #pragma unroll
    for (int i = 0; i < 6; ++i) {
        hz[i] = z[4 + i];
        o[4 + i] = s2 * g[PA[i]] * g[PB[i]] + s1 * hz[i];
    }
    #pragma unroll
    for (int j = 0; j < 10; ++j) {
        int a = TA[j], b = TB[j], c = TC[j];
        o[10 + j] = s3 * g[a] * g[b] * g[c]
                  + s2 * (hz[HIDX[a][b]] * g[c] + hz[HIDX[a][c]] * g[b] + hz[HIDX[b][c]] * g[a])
                  + s1 * z[10 + j];
    }
}

// in-place pointwise tanh jet update over all (point, neuron) sites
__device__ __forceinline__ void activation_pass(float* h, int tid) {
    for (int s = tid; s < NPTS * WIDTH; s += 256) {
        int p = s >> 7, n = s & 127;
        float z[NCH], o[NCH];
        #pragma unroll
        for (int c = 0; c < NCH; ++c) z[c] = h[(c * NPTS + p) * HP + n];
        tanh_chain(z, o);
        #pragma unroll
        for (int c = 0; c < NCH; ++c) h[(c * NPTS + p) * HP + n] = o[c];
    }
}

__global__ __launch_bounds__(256) void pinn_jet_kernel(Params P) {
    extern __shared__ float lds[];
    float* h  = lds + H_OFF;   // [NCH*NPTS][HP] jet state (k-major rows)
    float* wl = lds + W_OFF;   // [K][128] staged weights
    float* zf = lds + ZF_OFF;  // [NPTS][NCH] final-layer pre-activations

    const int tid  = threadIdx.x;
    const int lane = tid & 31;
    const int wave = tid >> 5;       // 0..7 -> 16-neuron output slice
    const int ln   = lane & 15;      // M row (A) / N col (B,C)
    const int hi   = lane >> 4;      // half-wave selector (K pairs / +8 rows)
    const int n0   = wave * 16;
    const int p0   = blockIdx.x * NPTS;

    // ---- init layer-0 jet: h[c][p][k<4]; value=(x,y,t,0), grad=I, higher=0
    for (int idx = tid; idx < NCH * NPTS * 4; idx += 256) {
        int k = idx & 3;
        int p = (idx >> 2) & 15;
        int c = idx >> 6;
        float v = 0.0f;
        if (p0 + p < P.n) {
            if (c == 0) {
                if (k == 0) v = P.x[p0 + p];
                else if (k == 1) v = P.y[p0 + p];
                else if (k == 2) v = P.t[p0 + p];
            } else if (c == 1 && k == 0) v = 1.0f;
            else if (c == 2 && k == 1) v = 1.0f;
            else if (c == 3 && k == 2) v = 1.0f;
        }
        h[(c * NPTS + p) * HP + k] = v;
    }

    // =========================== layer 0 (K = 4, zero-padded from 3) ==========
    {
        __syncthreads();
        for (int i = tid; i < 4 * WIDTH; i += 256)
            wl[i] = (i < 3 * WIDTH) ? P.W[0][i] : 0.0f;
        // prefetch next layer's weights into L2 while layer 0 runs
        __builtin_prefetch(P.W[1] + (tid << 6), 0, 0);
        __syncthreads();

        const float bn = P.b[0][n0 + ln];
        v2f b0;
        b0.x = wl[(2 * hi) * WIDTH + n0 + ln];
        b0.y = wl[(2 * hi + 1) * WIDTH + n0 + ln];

        // channel 0 peeled: bias folded into the WMMA C-operand (constant per lane)
        {
            v8f acc;
            #pragma unroll
            for (int v = 0; v < 8; ++v) acc[v] = bn;
            v2f a = *(const v2f*)&h[(0 * NPTS + ln) * HP + 2 * hi];
            acc = __builtin_amdgcn_wmma_f32_16x16x4_f32(
                false, a, false, b0, (short)0, acc, false, false);
            __syncthreads();                          // all reads of ch-0 rows done
            float* hw = &h[0 + n0 + ln];
            #pragma unroll
            for (int v = 0; v < 8; ++v)
                hw[(v + 8 * hi) * HP] = acc[v];       // C layout: M = v + 8*(lane/16)
        }
        for (int c = 1; c < NCH; ++c) {
            v8f acc = {};
            v2f a = *(const v2f*)&h[(c * NPTS + ln) * HP + 2 * hi];
            acc = __builtin_amdgcn_wmma_f32_16x16x4_f32(
                false, a, false, b0, (short)0, acc, false, false);
            __syncthreads();                          // all reads of ch-c rows done
            float* hw = &h[(c * NPTS) * HP + n0 + ln];
            #pragma unroll
            for (int v = 0; v < 8; ++v)
                hw[(v + 8 * hi) * HP] = acc[v];
        }
        __syncthreads();
        activation_pass(h, tid);
    }

    // =========================== layers 1..7 (K = 128, KT = 32) ===============
    for (int l = 1; l < 8; ++l) {
        __syncthreads();                              // prev writes visible, wl free
        for (int i = tid; i < WIDTH * WIDTH; i += 256)
            wl[i] = P.W[l][i];
        if (l < 7)                                    // hide next 64KB weight fetch
            __builtin_prefetch(P.W[l + 1] + (tid << 6), 0, 0);
        __syncthreads();

        const float bn = P.b[l][n0 + ln];

        // hoist this wave's full B slice (K=128 x 16 neurons) into 64 VGPRs
        v2f br[32];
        const float* wb = &wl[(2 * hi) * WIDTH + n0 + ln];
        #pragma unroll
        for (int kt = 0; kt < 32; ++kt) {
            br[kt].x = wb[kt * 4 * WIDTH];            // K = 4*kt + 2*hi
            br[kt].y = wb[kt * 4 * WIDTH + WIDTH];    // K = 4*kt + 2*hi + 1
        }

        // channel 0 peeled: bias folded into the accumulator init
        {
            v8f acc;
            #pragma unroll
            for (int v = 0; v < 8; ++v) acc[v] = bn;
            const float* ha = &h[(0 * NPTS + ln) * HP + 2 * hi];
            #pragma unroll
            for (int kt = 0; kt < 32; ++kt) {
                v2f a = *(const v2f*)(ha + kt * 4);
                acc = __builtin_amdgcn_wmma_f32_16x16x4_f32(
                    false, a, false, br[kt], (short)0, acc, false, false);
            }
            __syncthreads();
            float* hw = &h[0 + n0 + ln];
            #pragma unroll
            for (int v = 0; v < 8; ++v)
                hw[(v + 8 * hi) * HP] = acc[v];
        }
        for (int c = 1; c < NCH; ++c) {
            v8f acc = {};
            const float* ha = &h[(c * NPTS + ln) * HP + 2 * hi];
            #pragma unroll
            for (int kt = 0; kt < 32; ++kt) {
                v2f a = *(const v2f*)(ha + kt * 4);   // A: M=ln, K=4*kt+2*hi+V
                acc = __builtin_amdgcn_wmma_f32_16x16x4_f32(
                    false, a, false, br[kt], (short)0, acc, false, false);
            }
            __syncthreads();                          // all reads of ch-c rows done
            float* hw = &h[(c * NPTS) * HP + n0 + ln];
            #pragma unroll
            for (int v = 0; v < 8; ++v)
                hw[(v + 8 * hi) * HP] = acc[v];
        }
        __syncthreads();
        activation_pass(h, tid);
    }
    __syncthreads();

    // ---- final layer: psi = tanh-chain of column 0 of W8 (128x2)
    for (int i = tid; i < WIDTH; i += 256) wl[i] = P.W[8][2 * i];
    __syncthreads();
    for (int idx = tid; idx < NPTS * NCH; idx += 256) {
        int p = idx / NCH, c = idx % NCH;
        float sum = (c == 0) ? P.b[8][0] : 0.0f;
        const float* hr = &h[(c * NPTS + p) * HP];
        for (int k = 0; k < WIDTH; ++k) sum += hr[k] * wl[k];
        zf[p * NCH + c] = sum;
    }
    __syncthreads();

    if (tid < NPTS && p0 + tid < P.n) {
        float z[NCH], o[NCH];
        #pragma unroll
        for (int c = 0; c < NCH; ++c) z[c] = zf[tid * NCH + c];
        tanh_chain(z, o);
        // o: [psi, g0,g1,g2, H00,H01,H02,H11,H12,H22, T000,T001,T002,T011,T012,T022,T111,T112,T122,T222]
        float u   =  o[2];             // psi_y
        float v   = -o[1];             // -psi_x
        float u_x =  o[5],  u_y =  o[7],  u_t =  o[8];
        float v_x = -o[4],  v_y = -o[5],  v_t = -o[6];
        float u_xx =  o[11], u_yy =  o[16];
        float v_xx = -o[10], v_yy = -o[13];
        float l1 = P.lam1[0], l2 = P.lam2[0];
        float fu = u_t + l1 * (u * u_x + v * u_y) - l2 * (u_xx + u_yy);
        float fv = v_t + l1 * (u * v_x + v * v_y) - l2 * (v_xx + v_yy);
        int gp = p0 + tid;
        P.out[gp]           = u;
        P.out[P.n + gp]     = v;
        P.out[2 * P.n + gp] = fu;
        P.out[3 * P.n + gp] = fv;
    }
}

extern "C" void kernel_launch(void* const* d_in, const int* in_sizes, int n_in,
                              void* d_out, int out_size, void* d_ws, size_t ws_size,
                              hipStream_t stream) {
    Params P;
    P.x = (const float*)d_in[0];
    P.y = (const float*)d_in[1];
    P.t = (const float*)d_in[2];
    for (int i = 0; i < 9; ++i) {
        P.W[i] = (const float*)d_in[3 + 2 * i];
        P.b[i] = (const float*)d_in[4 + 2 * i];
    }
    P.lam1 = (const float*)d_in[21];
    P.lam2 = (const float*)d_in[22];
    P.out  = (float*)d_out;
    P.n    = in_sizes[0];

    int nblocks = (P.n + NPTS - 1) / NPTS;           // 2048 for N=32768
    size_t ldsbytes = LDS_FLOATS * sizeof(float);    // ~230 KB of CDNA5's 320 KB/WGP
    pinn_jet_kernel<<<nblocks, 256, ldsbytes, stream>>>(P);
}